// GraphInformationBottleneckModule_80779744903957
// MI455X (gfx1250) — compile-verified
//
#include <hip/hip_runtime.h>
#include <hip/hip_bf16.h>
#include <stdint.h>

// ---------------------------------------------------------------------------
// Problem constants (from reference)
// ---------------------------------------------------------------------------
#define NN 524288LL   // nodes
#define BB 16384      // graphs
#define DD 64         // feature dim
#define S2S_STEPS 2

typedef __attribute__((ext_vector_type(2))) float v2f;
typedef __attribute__((ext_vector_type(8))) float v8f;

// Only check for the WMMA builtin on the DEVICE pass (host clang lacks it).
#if defined(__gfx1250__) && !__has_builtin(__builtin_amdgcn_wmma_f32_16x16x4_f32)
#error "wmma_f32_16x16x4_f32 builtin not available on gfx1250 device pass"
#endif

// ---------------------------------------------------------------------------
// Helpers
// ---------------------------------------------------------------------------
__device__ __forceinline__ float sigm(float x) { return 1.0f / (1.0f + __expf(-x)); }

// monotone float <-> uint key for atomic max on signed floats
__device__ __forceinline__ unsigned f2key(float f) {
    unsigned u = __float_as_uint(f);
    return (u & 0x80000000u) ? ~u : (u | 0x80000000u);
}
__device__ __forceinline__ float key2f(unsigned k) {
    unsigned u = (k & 0x80000000u) ? (k & 0x7fffffffu) : ~k;
    return __uint_as_float(u);
}

// ---------------------------------------------------------------------------
// Generic fill
// ---------------------------------------------------------------------------
__global__ void k_fill(unsigned* __restrict__ p, unsigned v, size_t n) {
    size_t i = (size_t)blockIdx.x * blockDim.x + threadIdx.x;
    size_t stride = (size_t)gridDim.x * blockDim.x;
    for (; i < n; i += stride) p[i] = v;
}

// ---------------------------------------------------------------------------
// WMMA GEMM:  Y[M,Ncols] = act( A[M,K](lda) @ W[Ncols,K](ldw)^T + bias )
// grid = (M/128, Ncols/64), block = 256 (8 waves).
// Each wave computes a 16-row x 64-col strip: 4 accumulators share one A-frag
// per k-step (4x A reuse, 4 WMMAs per A load).
// f32 16x16x4 WMMA; K multiple of 4; M multiple of 128; Ncols multiple of 64.
// A-frag: lane l, elem v -> A[row=l&15][k + (l>>4)*2 + v]
// B-frag: lane l, elem v -> W[col=l&15][k + (l>>4)*2 + v]   (B[k][col] = W[col][k])
// C-frag: vgpr r, lane l -> Y[m0 + r + (l>>4)*8][j + (l&15)]
// ---------------------------------------------------------------------------
__global__ __launch_bounds__(256) void k_gemm_wmma(
    const float* __restrict__ A, int lda,
    const float* __restrict__ W, int ldw,
    const float* __restrict__ bias,
    float* __restrict__ Y, int ldy,
    int K, int act)
{
    const int lane = threadIdx.x & 31;
    const int wave = threadIdx.x >> 5;
    const int m0 = (blockIdx.x * 8 + wave) * 16;
    const int j0 = blockIdx.y * 64;
    const int lanelo = lane & 15;
    const int koff = (lane >> 4) * 2;

    const float* ap  = A + (size_t)(m0 + lanelo) * lda + koff;
    const float* bp0 = W + (size_t)(j0 + lanelo) * ldw + koff;
    const float* bp1 = bp0 + (size_t)16 * ldw;
    const float* bp2 = bp0 + (size_t)32 * ldw;
    const float* bp3 = bp0 + (size_t)48 * ldw;

    v8f c0 = {}, c1 = {}, c2 = {}, c3 = {};
#if defined(__gfx1250__)
    for (int k = 0; k < K; k += 4) {
        __builtin_prefetch(ap + k + 32, 0, 3);   // speculative A prefetch ahead
        v2f a, b0, b1, b2, b3;
        a.x  = ap[k];   a.y  = ap[k + 1];
        b0.x = bp0[k];  b0.y = bp0[k + 1];
        b1.x = bp1[k];  b1.y = bp1[k + 1];
        b2.x = bp2[k];  b2.y = bp2[k + 1];
        b3.x = bp3[k];  b3.y = bp3[k + 1];
        c0 = __builtin_amdgcn_wmma_f32_16x16x4_f32(false, a, false, b0, (short)0, c0, false, false);
        c1 = __builtin_amdgcn_wmma_f32_16x16x4_f32(false, a, false, b1, (short)0, c1, false, false);
        c2 = __builtin_amdgcn_wmma_f32_16x16x4_f32(false, a, false, b2, (short)0, c2, false, false);
        c3 = __builtin_amdgcn_wmma_f32_16x16x4_f32(false, a, false, b3, (short)0, c3, false, false);
    }
#endif

    const int rbase = m0 + ((lane >> 4) << 3);

#define STORE_TILE(creg, toff)                                              \
    do {                                                                    \
        float bj = bias[j0 + (toff) + lanelo];                              \
        _Pragma("unroll")                                                   \
        for (int r = 0; r < 8; ++r) {                                       \
            float v = creg[r] + bj;                                         \
            if (act) v = fmaxf(v, 0.0f);                                    \
            Y[(size_t)(rbase + r) * ldy + j0 + (toff) + lanelo] = v;        \
        }                                                                   \
    } while (0)

    STORE_TILE(c0, 0);
    STORE_TILE(c1, 16);
    STORE_TILE(c2, 32);
    STORE_TILE(c3, 48);
#undef STORE_TILE
}

// ---------------------------------------------------------------------------
// BatchNorm column stats over h[N,64]
// ---------------------------------------------------------------------------
__global__ void k_colstats(const float* __restrict__ h, long long nrows,
                           float* __restrict__ colsum, float* __restrict__ colsq)
{
    __shared__ float ss[256];
    __shared__ float sq[256];
    const int t = threadIdx.x;
    const int col = t & 63;
    const int rg = t >> 6;
    float s = 0.0f, q = 0.0f;
    for (long long r = (long long)blockIdx.x * 4 + rg; r < nrows;
         r += (long long)gridDim.x * 4) {
        float v = h[r * DD + col];
        s += v; q += v * v;
    }
    ss[t] = s; sq[t] = q;
    __syncthreads();
    if (t < 64) {
        s = ss[t] + ss[t + 64] + ss[t + 128] + ss[t + 192];
        q = sq[t] + sq[t + 64] + sq[t + 128] + sq[t + 192];
        atomicAdd(&colsum[col], s);
        atomicAdd(&colsq[col], q);
    }
}

__global__ void k_bn_finalize(const float* __restrict__ colsum,
                              const float* __restrict__ colsq,
                              float* __restrict__ muA, float* __restrict__ rstdA)
{
    int t = threadIdx.x;
    if (t < 64) {
        float inv_n = 1.0f / (float)NN;
        float mu = colsum[t] * inv_n;
        float var = colsq[t] * inv_n - mu * mu;   // biased var (training BN)
        muA[t] = mu;
        rstdA[t] = rsqrtf(var + 1e-5f);
    }
}

// ---------------------------------------------------------------------------
// Fused BN + ReLU + compressor L2 + concrete gate; lambda_pos out; preserve acc
// one thread per node
// ---------------------------------------------------------------------------
__global__ __launch_bounds__(256) void k_gate(
    const float* __restrict__ h,
    const float* __restrict__ muA, const float* __restrict__ rstdA,
    const float* __restrict__ bn_g, const float* __restrict__ bn_b,
    const float* __restrict__ c_w2, const float* __restrict__ c_b2,
    const float* __restrict__ eps_gate,
    float* __restrict__ lambda_out, float* __restrict__ preserve_acc)
{
    const long long i = (long long)blockIdx.x * 256 + threadIdx.x;
    const float* hr = h + i * DD;
    float p = c_b2[0];
#pragma unroll 8
    for (int j = 0; j < DD; ++j) {
        float v = (hr[j] - muA[j]) * rstdA[j] * bn_g[j] + bn_b[j];
        p += fmaxf(v, 0.0f) * c_w2[j];
    }
    const float bia = 1e-4f;
    float eg = eps_gate[i];
    float eps = (bia - (1.0f - bia)) * eg + (1.0f - bia);
    float gi = __logf(eps) - __logf(1.0f - eps);
    lambda_out[i] = sigm(gi + p);

    __shared__ float red[256];
    red[threadIdx.x] = (p > 0.0f) ? 1.0f : 0.0f;
    __syncthreads();
    for (int s = 128; s > 0; s >>= 1) {
        if (threadIdx.x < s) red[threadIdx.x] += red[threadIdx.x + s];
        __syncthreads();
    }
    if (threadIdx.x == 0) atomicAdd(preserve_acc, red[0]);
}

// ---------------------------------------------------------------------------
// Segment count + feature sum.  thread per (node, 4-dim group)
// ---------------------------------------------------------------------------
__global__ void k_seg_count_sum(const float* __restrict__ features,
                                const int* __restrict__ idx,
                                float* __restrict__ counts, float* __restrict__ sumF)
{
    const long long gid = (long long)blockIdx.x * 256 + threadIdx.x;
    const long long n = gid >> 4;
    const int q = (int)(gid & 15);
    const int b = idx[n];
    if (q == 0) atomicAdd(&counts[b], 1.0f);
    const float* fp = features + n * DD + q * 4;
    float* sp = sumF + (size_t)b * DD + q * 4;
#pragma unroll
    for (int k = 0; k < 4; ++k) atomicAdd(&sp[k], fp[k]);
}

__global__ void k_seg_mean(float* __restrict__ sumF, const float* __restrict__ counts)
{
    const long long gid = (long long)blockIdx.x * 256 + threadIdx.x;
    const long long b = gid >> 6;
    sumF[gid] /= fmaxf(counts[b], 1.0f);
}

__global__ void k_seg_var(const float* __restrict__ features,
                          const int* __restrict__ idx,
                          const float* __restrict__ meanG, float* __restrict__ varS)
{
    const long long gid = (long long)blockIdx.x * 256 + threadIdx.x;
    const long long n = gid >> 4;
    const int q = (int)(gid & 15);
    const int b = idx[n];
    const float* fp = features + n * DD + q * 4;
    const float* mp = meanG + (size_t)b * DD + q * 4;
    float* vp = varS + (size_t)b * DD + q * 4;
#pragma unroll
    for (int k = 0; k < 4; ++k) {
        float d = fp[k] - mp[k];
        atomicAdd(&vp[k], d * d);
    }
}

__global__ void k_seg_std(float* __restrict__ varS, const float* __restrict__ counts)
{
    const long long gid = (long long)blockIdx.x * 256 + threadIdx.x;
    const long long b = gid >> 6;
    varS[gid] = sqrtf(varS[gid] / fmaxf(counts[b] - 1.0f, 1.0f));
}

// ---------------------------------------------------------------------------
// noisy = lam*f + (1-lam)*mean + noise*(1-lam)*std  (overwrites h buffer)
// plus fused KL scalar accumulators.  thread per (node, 4-dim group)
// ---------------------------------------------------------------------------
__global__ __launch_bounds__(256) void k_noisy_kl(
    const float* __restrict__ features, const float* __restrict__ noise,
    const float* __restrict__ lambda, const float* __restrict__ meanG,
    const float* __restrict__ stdG, const int* __restrict__ idx,
    float* __restrict__ noisy, float* __restrict__ s1_acc, float* __restrict__ s2_acc)
{
    const long long gid = (long long)blockIdx.x * 256 + threadIdx.x;
    const long long n = gid >> 4;
    const int q = (int)(gid & 15);
    const int b = idx[n];
    const float lam = lambda[n];
    const float ln = 1.0f - lam;
    const float epsl = 1e-7f;

    const float* fp = features + n * DD + q * 4;
    const float* zp = noise + n * DD + q * 4;
    const float* mp = meanG + (size_t)b * DD + q * 4;
    const float* sp = stdG + (size_t)b * DD + q * 4;
    float* op = noisy + n * DD + q * 4;

    float kl1 = 0.0f, kl2 = 0.0f;
#pragma unroll
    for (int k = 0; k < 4; ++k) {
        float f = fp[k], m = mp[k], s = sp[k];
        float nmean = lam * f + ln * m;
        float nstd = ln * s;
        op[k] = nmean + zp[k] * nstd;
        float inv = 1.0f / (s + epsl);
        float a = nstd * inv;           // noisy_std / (std+eps)
        float c = (nmean - m) * inv;    // (noisy_mean-mean)/(std+eps)
        kl1 += a * a;
        kl2 += c * c;
    }
    kl1 *= 0.5f / (float)DD;

    __shared__ float r1[256];
    __shared__ float r2[256];
    r1[threadIdx.x] = kl1; r2[threadIdx.x] = kl2;
    __syncthreads();
    for (int s = 128; s > 0; s >>= 1) {
        if (threadIdx.x < s) {
            r1[threadIdx.x] += r1[threadIdx.x + s];
            r2[threadIdx.x] += r2[threadIdx.x + s];
        }
        __syncthreads();
    }
    if (threadIdx.x == 0) {
        atomicAdd(s1_acc, r1[0]);
        atomicAdd(s2_acc, r2[0]);
    }
}

// ---------------------------------------------------------------------------
// Set2Set pieces
// ---------------------------------------------------------------------------
__global__ void k_build_wcat(const float* __restrict__ wih, const float* __restrict__ whh,
                             const float* __restrict__ bih, const float* __restrict__ bhh,
                             float* __restrict__ Wcat, float* __restrict__ bcat)
{
    const int gid = blockIdx.x * 256 + threadIdx.x;
    if (gid < 256 * 192) {
        const int j = gid / 192;
        const int k = gid % 192;
        Wcat[gid] = (k < 128) ? wih[j * 128 + k] : whh[j * 64 + (k - 128)];
    }
    if (gid < 256) bcat[gid] = bih[gid] + bhh[gid];
}

__global__ void k_lstm_cell(const float* __restrict__ gates,
                            float* __restrict__ cc, float* __restrict__ hh,
                            float* __restrict__ xcat)
{
    const long long gid = (long long)blockIdx.x * 256 + threadIdx.x;
    const long long b = gid >> 6;
    const int d = (int)(gid & 63);
    const float* g = gates + b * 256;
    float gi = g[d], gf = g[64 + d], gg = g[128 + d], go = g[192 + d];
    float c = sigm(gf) * cc[gid] + sigm(gi) * tanhf(gg);
    cc[gid] = c;
    float h = sigm(go) * tanhf(c);
    hh[gid] = h;
    xcat[b * 192 + 128 + d] = h;    // LSTM hidden input for next step
}

__global__ void k_attn_e(const float* __restrict__ noisy, const float* __restrict__ hh,
                         const int* __restrict__ idx,
                         float* __restrict__ ebuf, unsigned* __restrict__ mkey)
{
    const long long n = (long long)blockIdx.x * 256 + threadIdx.x;
    const int b = idx[n];
    const float* nv = noisy + n * DD;
    const float* qv = hh + (size_t)b * DD;
    float e = 0.0f;
#pragma unroll 8
    for (int j = 0; j < DD; ++j) e += nv[j] * qv[j];
    ebuf[n] = e;
    atomicMax(&mkey[b], f2key(e));
}

__global__ void k_attn_w(float* __restrict__ ebuf, const unsigned* __restrict__ mkey,
                         const int* __restrict__ idx, float* __restrict__ denom)
{
    const long long n = (long long)blockIdx.x * 256 + threadIdx.x;
    const int b = idx[n];
    float m = key2f(mkey[b]);
    float w = __expf(ebuf[n] - m);
    ebuf[n] = w;
    atomicAdd(&denom[b], w);
}

__global__ void k_attn_r(const float* __restrict__ noisy, const float* __restrict__ wbuf,
                         const float* __restrict__ denom, const int* __restrict__ idx,
                         float* __restrict__ rbuf)
{
    const long long gid = (long long)blockIdx.x * 256 + threadIdx.x;
    const long long n = gid >> 4;
    const int q = (int)(gid & 15);
    const int b = idx[n];
    const float a = wbuf[n] / denom[b];
    const float* np = noisy + n * DD + q * 4;
    float* rp = rbuf + (size_t)b * DD + q * 4;
#pragma unroll
    for (int k = 0; k < 4; ++k) atomicAdd(&rp[k], a * np[k]);
}

__global__ void k_qstar(const float* __restrict__ hh, const float* __restrict__ rbuf,
                        float* __restrict__ xcat)
{
    const long long gid = (long long)blockIdx.x * 256 + threadIdx.x;
    const long long b = gid >> 6;
    const int d = (int)(gid & 63);
    xcat[b * 192 + d] = hh[gid];
    xcat[b * 192 + 64 + d] = rbuf[gid];
}

__global__ void k_finalize(float* __restrict__ out_kl, float* __restrict__ out_pr,
                           const float* __restrict__ scal)
{
    out_kl[0] = scal[0] / (float)BB + scal[1] / ((float)BB * (float)DD);
    out_pr[0] = scal[2] / (float)NN;
}

// ---------------------------------------------------------------------------
// Host driver
// ---------------------------------------------------------------------------
extern "C" void kernel_launch(void* const* d_in, const int* in_sizes, int n_in,
                              void* d_out, int out_size, void* d_ws, size_t ws_size,
                              hipStream_t stream) {
    (void)in_sizes; (void)n_in; (void)out_size; (void)ws_size;

    const float* features = (const float*)d_in[0];
    const int* idx        = (const int*)d_in[1];
    const float* eps_gate = (const float*)d_in[2];
    const float* noise    = (const float*)d_in[3];
    const float* c_w1     = (const float*)d_in[4];
    const float* c_b1     = (const float*)d_in[5];
    const float* bn_g     = (const float*)d_in[6];
    const float* bn_b     = (const float*)d_in[7];
    const float* c_w2     = (const float*)d_in[8];
    const float* c_b2     = (const float*)d_in[9];
    const float* lstm_wih = (const float*)d_in[10];
    const float* lstm_whh = (const float*)d_in[11];
    const float* lstm_bih = (const float*)d_in[12];
    const float* lstm_bhh = (const float*)d_in[13];
    const float* p_w1 = (const float*)d_in[14];
    const float* p_b1 = (const float*)d_in[15];
    const float* p_w2 = (const float*)d_in[16];
    const float* p_b2 = (const float*)d_in[17];
    const float* p_w3 = (const float*)d_in[18];
    const float* p_b3 = (const float*)d_in[19];

    float* out = (float*)d_out;
    float* out_pred = out;                               // [B,64]
    float* out_kl   = out + (size_t)BB * 64;             // scalar
    float* out_pr   = out_kl + 1;                        // scalar
    float* out_lam  = out_kl + 2;                        // [N]

    // workspace layout (floats)
    float* w = (float*)d_ws;
    size_t o = 0;
    float* hbuf  = w + o; o += (size_t)NN * DD;    // h, later noisy
    float* gates = w + o; o += (size_t)BB * 256;   // also predictor z1
    float* xcat  = w + o; o += (size_t)BB * 192;   // [q|r|h], also z2
    float* hh    = w + o; o += (size_t)BB * DD;
    float* cc    = w + o; o += (size_t)BB * DD;
    float* rbuf  = w + o; o += (size_t)BB * DD;
    float* sumF  = w + o; o += (size_t)BB * DD;    // -> mean_g
    float* varS  = w + o; o += (size_t)BB * DD;    // -> std_g
    float* ebuf  = w + o; o += (size_t)NN;         // e, then w
    float* counts = w + o; o += BB;
    float* denom  = w + o; o += BB;
    unsigned* mkey = (unsigned*)(w + o); o += BB;
    float* Wcat = w + o; o += 256 * 192;
    float* bcat = w + o; o += 256;
    float* colsum = w + o; o += 64;
    float* colsq  = w + o; o += 64;
    float* muA    = w + o; o += 64;
    float* rstdA  = w + o; o += 64;
    float* scal   = w + o; o += 8;   // [0]=S1 [1]=S2 [2]=preserve_count

    auto fill = [&](void* p, unsigned v, size_t n) {
        size_t blocks = (n + 255) / 256;
        if (blocks > 4096) blocks = 4096;
        k_fill<<<dim3((unsigned)blocks), dim3(256), 0, stream>>>((unsigned*)p, v, n);
    };

    // ---- init accumulators / state ----
    fill(xcat, 0, (size_t)BB * 192);
    fill(hh, 0, (size_t)BB * DD);
    fill(cc, 0, (size_t)BB * DD);
    fill(counts, 0, BB);
    fill(sumF, 0, (size_t)BB * DD);
    fill(varS, 0, (size_t)BB * DD);
    fill(colsum, 0, 256);   // colsum, colsq, muA, rstdA contiguous
    fill(scal, 0, 8);

    k_build_wcat<<<dim3((256 * 192 + 255) / 256), dim3(256), 0, stream>>>(
        lstm_wih, lstm_whh, lstm_bih, lstm_bhh, Wcat, bcat);

    // ---- compressor GEMM: h = features @ c_w1^T + c_b1 ----
    k_gemm_wmma<<<dim3((unsigned)(NN / 128), 1), dim3(256), 0, stream>>>(
        features, DD, c_w1, DD, c_b1, hbuf, DD, DD, 0);

    // ---- BN stats + finalize ----
    k_colstats<<<dim3(1024), dim3(256), 0, stream>>>(hbuf, NN, colsum, colsq);
    k_bn_finalize<<<dim3(1), dim3(64), 0, stream>>>(colsum, colsq, muA, rstdA);

    // ---- gate: BN+ReLU+Linear(64->1)+concrete gate -> lambda, preserve ----
    k_gate<<<dim3((unsigned)(NN / 256)), dim3(256), 0, stream>>>(
        hbuf, muA, rstdA, bn_g, bn_b, c_w2, c_b2, eps_gate, out_lam, scal + 2);

    // ---- per-graph mean / unbiased std ----
    k_seg_count_sum<<<dim3((unsigned)(NN * 16 / 256)), dim3(256), 0, stream>>>(
        features, idx, counts, sumF);
    k_seg_mean<<<dim3((unsigned)((size_t)BB * DD / 256)), dim3(256), 0, stream>>>(sumF, counts);
    k_seg_var<<<dim3((unsigned)(NN * 16 / 256)), dim3(256), 0, stream>>>(
        features, idx, sumF, varS);
    k_seg_std<<<dim3((unsigned)((size_t)BB * DD / 256)), dim3(256), 0, stream>>>(varS, counts);

    // ---- noisy (overwrites hbuf) + KL accumulators ----
    k_noisy_kl<<<dim3((unsigned)(NN * 16 / 256)), dim3(256), 0, stream>>>(
        features, noise, out_lam, sumF, varS, idx, hbuf, scal + 0, scal + 1);

    // ---- Set2Set ----
    for (int step = 0; step < S2S_STEPS; ++step) {
        fill(denom, 0, BB);
        fill(mkey, 0, BB);                 // key(x) >= 0 always; 0 = -inf sentinel
        fill(rbuf, 0, (size_t)BB * DD);

        // gates = [q_star|hh] @ Wcat^T + bcat
        k_gemm_wmma<<<dim3(BB / 128, 256 / 64), dim3(256), 0, stream>>>(
            xcat, 192, Wcat, 192, bcat, gates, 256, 192, 0);
        k_lstm_cell<<<dim3((unsigned)((size_t)BB * DD / 256)), dim3(256), 0, stream>>>(
            gates, cc, hh, xcat);

        k_attn_e<<<dim3((unsigned)(NN / 256)), dim3(256), 0, stream>>>(
            hbuf, hh, idx, ebuf, mkey);
        k_attn_w<<<dim3((unsigned)(NN / 256)), dim3(256), 0, stream>>>(
            ebuf, mkey, idx, denom);
        k_attn_r<<<dim3((unsigned)(NN * 16 / 256)), dim3(256), 0, stream>>>(
            hbuf, ebuf, denom, idx, rbuf);
        k_qstar<<<dim3((unsigned)((size_t)BB * DD / 256)), dim3(256), 0, stream>>>(
            hh, rbuf, xcat);
    }

    // ---- predictor MLP: 128 -> 256 -> 128 -> 64 ----
    // z1 = relu(q_star @ p_w1^T + p_b1): q_star = first 128 cols of xcat (lda=192)
    k_gemm_wmma<<<dim3(BB / 128, 256 / 64), dim3(256), 0, stream>>>(
        xcat, 192, p_w1, 128, p_b1, gates, 256, 128, 1);
    // z2 = relu(z1 @ p_w2^T + p_b2): overlay z2 on xcat region (ldy=128)
    k_gemm_wmma<<<dim3(BB / 128, 128 / 64), dim3(256), 0, stream>>>(
        gates, 256, p_w2, 256, p_b2, xcat, 128, 256, 1);
    // pred = z2 @ p_w3^T + p_b3 -> d_out
    k_gemm_wmma<<<dim3(BB / 128, 64 / 64), dim3(256), 0, stream>>>(
        xcat, 128, p_w3, 128, p_b3, out_pred, 64, 128, 0);

    k_finalize<<<dim3(1), dim3(1), 0, stream>>>(out_kl, out_pr, scal);
}